// NumProtoLoss_17858474017094
// MI455X (gfx1250) — compile-verified
//
#include <hip/hip_runtime.h>
#include <math.h>

// ---------------------------------------------------------------------------
// NumProtoLoss top-4 masking, MI455X (gfx1250).
//
// out[s,p,c] = in[s,p,c] unless p is one of the top-4 prototypes of column
// (s,c) (ties -> lower index, matching stable argsort(-x)), in which case 0.
//
// Memory-bound: 819 MB traffic -> ~35us floor @ 23.3 TB/s. We stream the
// tensor exactly once through a triple-buffered LDS pipeline fed by
// global_load_async_to_lds_b128 (ASYNCcnt), scan top-4 per column from LDS,
// write through with non-temporal stores, then scatter 4 zeros per column.
// ---------------------------------------------------------------------------

namespace {

constexpr int kS   = 256;
constexpr int kP   = 2000;
constexpr int kC   = 200;
constexpr int kPC  = kP * kC;               // 400000 elements per sample slab
constexpr int BLK  = 128;                   // threads per block == columns per block
constexpr int KP   = 40;                    // rows per LDS tile
constexpr int NT   = kP / KP;               // 50 tiles
constexpr int OPS  = (KP * BLK / 4) / BLK;  // 10 async b128 ops / thread / tile
constexpr int TILE_BYTES = KP * BLK * 4;    // 20480 B
constexpr int NBUF = 3;                     // triple buffer -> one barrier/tile

#define AS1 __attribute__((address_space(1)))
#define AS3 __attribute__((address_space(3)))

typedef int v4i __attribute__((ext_vector_type(4)));

__device__ __forceinline__ void async_load_b128(const float* g, void* lds_generic) {
#if __has_builtin(__builtin_amdgcn_global_load_async_to_lds_b128)
  // Global addresses are numerically identical in generic and AS(1) space on
  // AMDGPU; the integer round-trip also drops the const qualifier.
  AS1 v4i* gp = (AS1 v4i*)(unsigned long long)g;
  AS3 v4i* lp = (AS3 v4i*)lds_generic;      // true addrspacecast (real LDS ptr)
  __builtin_amdgcn_global_load_async_to_lds_b128(gp, lp, /*offset=*/0, /*cpol=*/0);
#else
  unsigned lofs = (unsigned)(unsigned long long)(AS3 v4i*)lds_generic;
  asm volatile("global_load_async_to_lds_b128 %0, %1, off"
               :: "v"(lofs), "v"(g) : "memory");
#endif
}

template <int N>
__device__ __forceinline__ void wait_asynccnt() {
#if __has_builtin(__builtin_amdgcn_s_wait_asynccnt)
  __builtin_amdgcn_s_wait_asynccnt(N);
#else
  asm volatile("s_wait_asynccnt %0" :: "i"(N) : "memory");
#endif
}

__global__ void __launch_bounds__(BLK)
numproto_top4_mask(const float* __restrict__ in, float* __restrict__ out) {
  __shared__ char smem[NBUF * TILE_BYTES];  // 60 KB

  const int tid = threadIdx.x;
  const int t   = blockIdx.x * BLK + tid;   // flat column id in [0, S*C)
  const int s   = t / kC;
  const int c   = t - s * kC;
  const int colbase = s * kPC + c;          // < 102.4M, fits int32

  // Per-thread async-copy descriptors; constant across tiles except row base.
  // Op o = i*BLK + tid covers row pp = o>>5, 16B group j = o&31 of the tile:
  // a wave's 32 lanes cover one contiguous 512B tile row. Groups of 4 columns
  // never straddle a sample slab (4 | 200 and 4 | blockStart) and are 16B
  // aligned in both global memory and LDS.
  unsigned gofs[OPS];
  unsigned lofs[OPS];
#pragma unroll
  for (int i = 0; i < OPS; ++i) {
    const int o    = i * BLK + tid;
    const int pp   = o >> 5;
    const int j    = o & 31;
    const int col0 = blockIdx.x * BLK + 4 * j;
    const int sq   = col0 / kC;
    const int cq   = col0 - sq * kC;
    gofs[i] = (unsigned)(sq * kPC + cq + pp * kC);   // element offset at p0=0
    lofs[i] = (unsigned)((pp * BLK + 4 * j) * 4);    // byte offset in tile
  }

  auto issue = [&](int k) {
    const unsigned buf  = (unsigned)((k % NBUF) * TILE_BYTES);
    const unsigned rofs = (unsigned)(k * KP * kC);   // row base in elements
#pragma unroll
    for (int i = 0; i < OPS; ++i) {
      async_load_b128(in + gofs[i] + rofs, smem + (lofs[i] + buf));
    }
  };

  issue(0);
  issue(1);

  // Top-4 (value desc, index asc). Scanning p ascending + strict '>' exactly
  // reproduces stable argsort(-x) tie-breaking.
  float v0 = -INFINITY, v1 = -INFINITY, v2 = -INFINITY, v3 = -INFINITY;
  int   i0 = 0, i1 = 0, i2 = 0, i3 = 0;

  for (int k = 0; k < NT; ++k) {
    // Async loads complete in order: <=OPS outstanding retires tile k while
    // tile k+1 stays in flight. Barrier makes the whole tile visible.
    if (k + 1 < NT) wait_asynccnt<OPS>();
    else            wait_asynccnt<0>();
    __syncthreads();

    const float* sb   = (const float*)(smem + (k % NBUF) * TILE_BYTES);
    const int    pbase = k * KP;
#pragma unroll
    for (int pp = 0; pp < KP; ++pp) {
      const float v = sb[pp * BLK + tid];            // ds_load, conflict-free
      const int   p = pbase + pp;
      __builtin_nontemporal_store(v, out + colbase + p * kC);  // write-once
      if (v > v3) {
        if (v > v2) {
          v3 = v2; i3 = i2;
          if (v > v1) {
            v2 = v1; i2 = i1;
            if (v > v0) { v1 = v0; i1 = i0; v0 = v; i0 = p; }
            else        { v1 = v;  i1 = p; }
          } else { v2 = v; i2 = p; }
        } else { v3 = v; i3 = p; }
      }
    }

    // buf[(k+2)%3]'s last readers finished scan(k-1) before this iteration's
    // barrier, so the prefetch is WAR-safe with a single barrier per tile.
    if (k + 2 < NT) issue(k + 2);
  }

  // Scatter zeros over the top-4 slots. Same-thread store-after-store to the
  // same address is ordered, so no fence is required.
  out[colbase + i0 * kC] = 0.0f;
  out[colbase + i1 * kC] = 0.0f;
  out[colbase + i2 * kC] = 0.0f;
  out[colbase + i3 * kC] = 0.0f;
}

}  // namespace

extern "C" void kernel_launch(void* const* d_in, const int* in_sizes, int n_in,
                              void* d_out, int out_size, void* d_ws, size_t ws_size,
                              hipStream_t stream) {
  (void)in_sizes; (void)n_in; (void)out_size; (void)d_ws; (void)ws_size;
  const float* in  = (const float*)d_in[0];
  float*       out = (float*)d_out;
  dim3 grid((kS * kC) / BLK);   // 400 blocks
  dim3 block(BLK);              // 128 threads = 4 wave32
  hipLaunchKernelGGL(numproto_top4_mask, grid, block, 0, stream, in, out);
}